// MultitaskPccRateDistortionLoss_11184094838809
// MI455X (gfx1250) — compile-verified
//
#include <hip/hip_runtime.h>
#include <hip/hip_bf16.h>
#include <math.h>

typedef __attribute__((ext_vector_type(2))) float v2f;
typedef __attribute__((ext_vector_type(8))) float v8f;

#define NUM_CLASSES 40

// ---------------------------------------------------------------------------
// D = A(16x4) * B(4x16) + 0, f32, one WMMA per 16x16 distance tile.
// ---------------------------------------------------------------------------
__device__ __forceinline__ v8f wmma_16x16x4_f32(v2f a, v2f b) {
  v8f c = {};
  // 8 args: (neg_a, A, neg_b, B, c_mod, C, reuse_a, reuse_b)
  return __builtin_amdgcn_wmma_f32_16x16x4_f32(false, a, false, b, (short)0, c,
                                               false, false);
}

// ---------------------------------------------------------------------------
// init: zero the 3 accumulators in workspace
// ws[0] = sum(log lik), ws[1] = cls mean accumulator, ws[2] = chamfer min-sum
// ---------------------------------------------------------------------------
__global__ void init_acc_kernel(float* acc) {
  if (threadIdx.x < 3) acc[threadIdx.x] = 0.0f;
}

// ---------------------------------------------------------------------------
// Chamfer half: for each point in X, min squared distance to all points in Y.
// One wave (32 threads) per 16-row tile; sweeps all P columns with WMMA.
// Accumulates sum of row-mins into acc via f32 global atomic add.
// ---------------------------------------------------------------------------
__global__ void __launch_bounds__(32)
chamfer_min_kernel(const float* __restrict__ X, const float* __restrict__ Y,
                   float* __restrict__ acc, int P) {
  const int lane = threadIdx.x;       // 0..31 (wave32)
  const int l16  = lane & 15;
  const int half = lane >> 4;
  const int n    = blockIdx.y;
  const int rbase = blockIdx.x * 16;

  const float* Xn = X + (size_t)n * P * 3;
  const float* Yn = Y + (size_t)n * P * 3;

  // A tile: row M = rbase + l16; lanes 0-15 hold K=0,1 ; lanes 16-31 K=2,pad.
  // Pre-scale by -2 so the WMMA directly produces -2*x.y.
  {
    const float* xp = Xn + (size_t)(rbase + l16) * 3;
    // loaded below into 'a'
  }
  const float* xp = Xn + (size_t)(rbase + l16) * 3;
  v2f a;
  if (half == 0) { a.x = xp[0]; a.y = xp[1]; }
  else           { a.x = xp[2]; a.y = 0.0f; }
  a.x *= -2.0f;
  a.y *= -2.0f;

  float rm[8];
#pragma unroll
  for (int r = 0; r < 8; ++r) rm[r] = __builtin_inff();

#pragma unroll 4
  for (int cbase = 0; cbase < P; cbase += 16) {
    const int col = cbase + l16;
    const float* yp = Yn + (size_t)col * 3;
    const float y0 = yp[0], y1 = yp[1], y2c = yp[2];
    const float ysq = y0 * y0 + y1 * y1 + y2c * y2c;

    // B tile (4x16, KxN): lane holds column N=l16; lanes 0-15 K=0,1 ;
    // lanes 16-31 K=2,pad.
    v2f b;
    if (half == 0) { b.x = y0;  b.y = y1;   }
    else           { b.x = y2c; b.y = 0.0f; }

    v8f c = wmma_16x16x4_f32(a, b);   // c[r] = -2 * x_{M} . y_{N}

#pragma unroll
    for (int r = 0; r < 8; ++r)
      rm[r] = fminf(rm[r], c[r] + ysq);   // min_j (y^2 - 2 x.y), partial
  }

  // Cross-lane min within each 16-lane half (masks < 16 keep halves apart),
  // then add x^2 of the owning row and locally sum the 16 row results.
  float localsum = 0.0f;
#pragma unroll
  for (int r = 0; r < 8; ++r) {
    float v = rm[r];
#pragma unroll
    for (int m = 8; m >= 1; m >>= 1)
      v = fminf(v, __shfl_xor(v, m, 32));
    if (l16 == 0) {
      const int row = rbase + r + 8 * half;   // C/D layout: VGPR r -> M=r(+8)
      const float* xq = Xn + (size_t)row * 3;
      const float xsq = xq[0] * xq[0] + xq[1] * xq[1] + xq[2] * xq[2];
      localsum += v + xsq;
    }
  }
  // combine lane0 (rows 0..7) with lane16 (rows 8..15)
  localsum += __shfl_xor(localsum, 16, 32);
  if (lane == 0) atomicAdd(acc, localsum);
}

// ---------------------------------------------------------------------------
// bpp: grid-stride sum of log(lik) with wave32 shuffle reduction.
// ---------------------------------------------------------------------------
__global__ void bpp_log_sum_kernel(const float* __restrict__ lik, long long n,
                                   float* __restrict__ acc) {
  long long i = (long long)blockIdx.x * blockDim.x + threadIdx.x;
  const long long stride = (long long)gridDim.x * blockDim.x;
  float s = 0.0f;
  for (; i < n; i += stride) s += logf(lik[i]);
#pragma unroll
  for (int m = 16; m >= 1; m >>= 1) s += __shfl_xor(s, m, 32);
  if ((threadIdx.x & 31) == 0) atomicAdd(acc, s);
}

// ---------------------------------------------------------------------------
// cls: per-row logsumexp cross-entropy, mean over N. One wave, row per lane.
// ---------------------------------------------------------------------------
__global__ void __launch_bounds__(32)
cls_kernel(const float* __restrict__ t, const int* __restrict__ label,
           float* __restrict__ acc, int N) {
  const int i = threadIdx.x;
  if (i < N) {
    const float* row = t + (size_t)i * NUM_CLASSES;
    float mx = -__builtin_inff();
    for (int c = 0; c < NUM_CLASSES; ++c) mx = fmaxf(mx, row[c]);
    float se = 0.0f;
    for (int c = 0; c < NUM_CLASSES; ++c) se += expf(row[c] - mx);
    const float lse = mx + logf(se);
    const int lb = label[i];
    const float nll = lse - row[lb];
    atomicAdd(acc + 1, nll / (float)N);
  }
}

// ---------------------------------------------------------------------------
// finalize: compose (loss, rec, cls, bpp)
// ---------------------------------------------------------------------------
__global__ void finalize_kernel(const float* __restrict__ acc,
                                float* __restrict__ out, int N, int P) {
  if (threadIdx.x == 0 && blockIdx.x == 0) {
    const float np = (float)N * (float)P;
    const float bpp = acc[0] * (-1.0f / (0.69314718055994531f * np));
    const float rec = acc[2] / np;  // (sum minx + sum miny) / (N*P)
    const float cls = acc[1];
    out[0] = rec + cls + bpp;
    out[1] = rec;
    out[2] = cls;
    out[3] = bpp;
  }
}

extern "C" void kernel_launch(void* const* d_in, const int* in_sizes, int n_in,
                              void* d_out, int out_size, void* d_ws,
                              size_t ws_size, hipStream_t stream) {
  const float* x_hat = (const float*)d_in[0];   // [N, P, 3]
  const float* pos   = (const float*)d_in[1];   // [N, P, 3]
  const float* t_hat = (const float*)d_in[2];   // [N, C]
  const int*   label = (const int*)d_in[3];     // [N, 1]
  const float* lik_y = (const float*)d_in[4];   // [N, 256, 1024]
  float* out = (float*)d_out;
  float* acc = (float*)d_ws;                    // acc[0]=bpp, [1]=cls, [2]=rec

  const int N = in_sizes[2] / NUM_CLASSES;      // 8
  const int P = in_sizes[0] / (3 * N);          // 4096
  const long long nlik = (long long)in_sizes[4];

  init_acc_kernel<<<1, 32, 0, stream>>>(acc);

  // bidirectional chamfer: two passes with roles swapped, both into acc[2]
  dim3 cgrid(P / 16, N);
  chamfer_min_kernel<<<cgrid, 32, 0, stream>>>(x_hat, pos, acc + 2, P);
  chamfer_min_kernel<<<cgrid, 32, 0, stream>>>(pos, x_hat, acc + 2, P);

  bpp_log_sum_kernel<<<1024, 256, 0, stream>>>(lik_y, nlik, acc);

  cls_kernel<<<1, 32, 0, stream>>>(t_hat, label, acc, N);

  finalize_kernel<<<1, 32, 0, stream>>>(acc, out, N, P);
}